// TemporalMotor_67963562492054
// MI455X (gfx1250) — compile-verified
//
#include <hip/hip_runtime.h>
#include <cstdint>
#include <cstddef>

typedef float v4f __attribute__((ext_vector_type(4)));

#define TPB        256
#define GROUPS     2          // vector groups (of 4 elements) per thread -> 2048 elems/block
#define SEG_STRIDE 20         // floats per segment record (5 x float4)

// CDNA5 async global->LDS load (tracked by ASYNCcnt, no VGPR staging).
__device__ __forceinline__ void async_ld_b32(uint32_t lds_off, uint32_t goff, const void* sbase) {
    asm volatile("global_load_async_to_lds_b32 %0, %1, %2"
                 :: "v"(lds_off), "v"(goff), "s"(sbase) : "memory");
}
__device__ __forceinline__ void wait_async0() {
    asm volatile("s_wait_asynccnt 0" ::: "memory");
}

struct Res { float t0, t1, t2; v4f q; };

// Hot path: binary search + 5x ds_load_b128 segment record + 2x v_sin_f32.
__device__ __forceinline__ Res eval_one(float tf,
                                        const float* __restrict__ s_times,
                                        const float* __restrict__ s_seg, int K)
{
    // searchsorted(keyframe_times, tf, side='left')
    int lo = 0, hi = K;
    while (lo < hi) {
        int mid = (lo + hi) >> 1;
        if (s_times[mid] < tf) lo = mid + 1; else hi = mid;
    }
    int idx = lo;
    if (idx < 1)     idx = 1;
    if (idx > K - 1) idx = K - 1;

    const v4f* R = (const v4f*)(s_seg + SEG_STRIDE * (idx - 1));
    v4f h0 = R[0];   // { t_prev, inv_dt, theta, rsin }
    v4f h1 = R[1];   // { trans_prev.xyz, ad }
    v4f h2 = R[2];   // { trans_next.xyz, 0 }
    v4f a  = R[3];   // quat_prev (normalized)
    v4f b  = R[4];   // quat_next (normalized, sign-flipped for shortest arc)

    float lt = (tf - h0.x) * h0.y;
    float w0 = 1.0f - lt;

    // slerp weights: sin((1-t)theta)/sin(theta), sin(t*theta)/sin(theta)
    float s0 = __sinf(w0 * h0.z) * h0.w;
    float s1 = __sinf(lt * h0.z) * h0.w;

    Res r;
    r.t0 = w0 * h1.x + lt * h2.x;
    r.t1 = w0 * h1.y + lt * h2.y;
    r.t2 = w0 * h1.z + lt * h2.z;

    v4f sl = s0 * a + s1 * b;

    // normalized-lerp fallback (uses flipped b, like the reference)
    v4f l  = a + lt * (b - a);
    float n2 = l.x * l.x + l.y * l.y + l.z * l.z + l.w * l.w;
    float ln = __builtin_amdgcn_sqrtf(n2);
    float li = __builtin_amdgcn_rcpf(fmaxf(ln, 1e-12f));

    bool nl = h1.w > 0.9995f;
    r.q.x = nl ? l.x * li : sl.x;
    r.q.y = nl ? l.y * li : sl.y;
    r.q.z = nl ? l.z * li : sl.z;
    r.q.w = nl ? l.w * li : sl.w;
    return r;
}

__global__ void __launch_bounds__(TPB)
TemporalMotor_kernel(const float* __restrict__ t,
                     const float* __restrict__ translations,   // (K,3)
                     const float* __restrict__ quaternions,    // (K,4)
                     const float* __restrict__ keyframe_times, // (K,)
                     float* __restrict__ out,                  // trans (N,3) then quat (N,4)
                     int N, int K)
{
    extern __shared__ __align__(16) float smem[];
    float* s_seg   = smem;                            // 20*(K-1) floats, 16B aligned
    float* s_times = s_seg + SEG_STRIDE * (K - 1);    // K
    float* s_trans = s_times + K;                     // 3K
    float* s_quat  = s_trans + 3 * K;                 // 4K

    const int tid = threadIdx.x;

    // ---- Stage keyframe tables into LDS with CDNA5 async copies (wave 0; wave32-uniform) ----
    if (tid < 32) {
        uint32_t off_t = (uint32_t)(uintptr_t)(void*)s_times;
        uint32_t off_r = (uint32_t)(uintptr_t)(void*)s_trans;
        uint32_t off_q = (uint32_t)(uintptr_t)(void*)s_quat;
        for (int j = tid; j < K;     j += 32) async_ld_b32(off_t + 4u * j, 4u * j, keyframe_times);
        for (int j = tid; j < 3 * K; j += 32) async_ld_b32(off_r + 4u * j, 4u * j, translations);
        for (int j = tid; j < 4 * K; j += 32) async_ld_b32(off_q + 4u * j, 4u * j, quaternions);
        wait_async0();
    }
    __syncthreads();

    // ---- Normalize quaternions once per block ----
    if (tid < K) {
        float q0 = s_quat[4 * tid + 0], q1 = s_quat[4 * tid + 1];
        float q2 = s_quat[4 * tid + 2], q3 = s_quat[4 * tid + 3];
        float n   = sqrtf(q0 * q0 + q1 * q1 + q2 * q2 + q3 * q3);
        float inv = 1.0f / fmaxf(n, 1e-12f);
        s_quat[4 * tid + 0] = q0 * inv;
        s_quat[4 * tid + 1] = q1 * inv;
        s_quat[4 * tid + 2] = q2 * inv;
        s_quat[4 * tid + 3] = q3 * inv;
    }
    __syncthreads();

    // ---- Build per-segment records: all segment-invariant slerp math done ONCE here ----
    if (tid < K - 1) {
        int j = tid;
        float a0 = s_quat[4 * j + 0], a1 = s_quat[4 * j + 1];
        float a2 = s_quat[4 * j + 2], a3 = s_quat[4 * j + 3];
        float b0 = s_quat[4 * j + 4], b1 = s_quat[4 * j + 5];
        float b2 = s_quat[4 * j + 6], b3 = s_quat[4 * j + 7];

        float dot = a0 * b0 + a1 * b1 + a2 * b2 + a3 * b3;
        float sg  = (dot < 0.0f) ? -1.0f : 1.0f;
        b0 *= sg; b1 *= sg; b2 *= sg; b3 *= sg;

        float ad    = fminf(fabsf(dot), 1.0f);
        float theta = acosf(ad);            // precise libm, once per segment
        float st    = sinf(theta);          // precise libm, once per segment
        float ssin  = (st < 1e-8f) ? 1.0f : st;
        float rsin  = 1.0f / ssin;

        float tp    = s_times[j], tn = s_times[j + 1];
        float invdt = 1.0f / (tn - tp + 1e-8f);

        float* r = s_seg + SEG_STRIDE * j;
        r[0]  = tp;                  r[1]  = invdt;
        r[2]  = theta;               r[3]  = rsin;
        r[4]  = s_trans[3 * j + 0];  r[5]  = s_trans[3 * j + 1];
        r[6]  = s_trans[3 * j + 2];  r[7]  = ad;
        r[8]  = s_trans[3 * j + 3];  r[9]  = s_trans[3 * j + 4];
        r[10] = s_trans[3 * j + 5];  r[11] = 0.0f;
        r[12] = a0; r[13] = a1; r[14] = a2; r[15] = a3;
        r[16] = b0; r[17] = b1; r[18] = b2; r[19] = b3;
    }
    __syncthreads();

    float* otr = out;
    float* oq  = out + 3ll * (long long)N;

    const bool vec_ok = ((N & 3) == 0) &&
                        ((((uintptr_t)t)   & 15) == 0) &&
                        ((((uintptr_t)otr) & 15) == 0) &&
                        ((((uintptr_t)oq)  & 15) == 0);

    const long long base = (long long)blockIdx.x * (TPB * GROUPS * 4);

    if (vec_ok) {
        // ---- fully b128 path: 1 NT load + 7 NT stores per 4 elements ----
        #pragma unroll
        for (int g = 0; g < GROUPS; ++g) {
            long long i0 = base + ((long long)g * TPB + tid) * 4;
            if (i0 >= N) break;
            v4f tv = __builtin_nontemporal_load((const v4f*)(t + i0));
            Res r0 = eval_one(tv.x, s_times, s_seg, K);
            Res r1 = eval_one(tv.y, s_times, s_seg, K);
            Res r2 = eval_one(tv.z, s_times, s_seg, K);
            Res r3 = eval_one(tv.w, s_times, s_seg, K);

            v4f T0 = { r0.t0, r0.t1, r0.t2, r1.t0 };
            v4f T1 = { r1.t1, r1.t2, r2.t0, r2.t1 };
            v4f T2 = { r2.t2, r3.t0, r3.t1, r3.t2 };
            v4f* tp4 = (v4f*)(otr + 3 * i0);
            __builtin_nontemporal_store(T0, tp4 + 0);
            __builtin_nontemporal_store(T1, tp4 + 1);
            __builtin_nontemporal_store(T2, tp4 + 2);

            v4f* qp4 = (v4f*)(oq + 4 * i0);
            __builtin_nontemporal_store(r0.q, qp4 + 0);
            __builtin_nontemporal_store(r1.q, qp4 + 1);
            __builtin_nontemporal_store(r2.q, qp4 + 2);
            __builtin_nontemporal_store(r3.q, qp4 + 3);
        }
    } else {
        // ---- scalar fallback (same coverage: TPB*GROUPS*4 elements per block) ----
        for (int k = 0; k < GROUPS * 4; ++k) {
            long long i = base + (long long)k * TPB + tid;
            if (i >= N) break;
            float tf = __builtin_nontemporal_load(&t[i]);
            Res r = eval_one(tf, s_times, s_seg, K);
            __builtin_nontemporal_store(r.t0,  &otr[3 * i + 0]);
            __builtin_nontemporal_store(r.t1,  &otr[3 * i + 1]);
            __builtin_nontemporal_store(r.t2,  &otr[3 * i + 2]);
            __builtin_nontemporal_store(r.q.x, &oq[4 * i + 0]);
            __builtin_nontemporal_store(r.q.y, &oq[4 * i + 1]);
            __builtin_nontemporal_store(r.q.z, &oq[4 * i + 2]);
            __builtin_nontemporal_store(r.q.w, &oq[4 * i + 3]);
        }
    }
}

extern "C" void kernel_launch(void* const* d_in, const int* in_sizes, int n_in,
                              void* d_out, int out_size, void* d_ws, size_t ws_size,
                              hipStream_t stream) {
    const float* t     = (const float*)d_in[0];
    const float* trans = (const float*)d_in[1];
    const float* quat  = (const float*)d_in[2];
    const float* times = (const float*)d_in[3];
    float* out = (float*)d_out;

    int N = in_sizes[0];       // number of time samples
    int K = in_sizes[3];       // number of keyframes

    if (N <= 0 || K < 2) return;

    const int per_block = TPB * GROUPS * 4;              // 2048 elements
    int blocks = (N + per_block - 1) / per_block;
    size_t shmem = (size_t)(SEG_STRIDE * (K - 1) + 8 * K) * sizeof(float);

    TemporalMotor_kernel<<<blocks, TPB, shmem, stream>>>(t, trans, quat, times, out, N, K);
}